// MultiHeadPool_27736898798338
// MI455X (gfx1250) — compile-verified
//
#include <hip/hip_runtime.h>
#include <hip/hip_bf16.h>
#include <cstddef>

// ---------------------------------------------------------------------------
// Problem constants
// ---------------------------------------------------------------------------
#define Bc   8
#define CM   256
#define Pn   16384
#define Hh   8
#define Ff   64
#define Gg   64
#define Oc   536          // H*(F+3)
#define OP   544          // Oc padded to multiple of 16 (34 tiles)
#define MT   34           // M tiles
#define NT   64           // points per workgroup
#define LDSW 65           // padded LDS row stride (bank-conflict free)
#define PT   (Pn / NT)    // 256 point-tiles per batch

typedef float v2f __attribute__((ext_vector_type(2)));
typedef float v8f __attribute__((ext_vector_type(8)));

// ---------------------------------------------------------------------------
// Compute one M-row of 4 N-tiles: C[mt*16+.., p0 + 0..63] with K = 256.
// A (16x4 f32): lanes 0-15 M=0..15 hold {K0,K1}; lanes 16-31 hold {K2,K3}.
// B (4x16 f32): mirrored — lanes 0-15 N=0..15 hold {K0,K1}; hi lanes {K2,K3}.
// WTi is packed as [k/4][m][4] so each lane's A-pair is one b64 load,
// reused across all 4 N-tiles (4 WMMAs per A load).
// ---------------------------------------------------------------------------
__device__ inline void wmma_row(const float* __restrict__ X,
                                const float* __restrict__ WTi,
                                int bb, int mt, int p0, v8f acc[4]) {
  const int lane = threadIdx.x & 31;
  const int half = lane >> 4;
  const int l16  = lane & 15;
  const float* xb = X + (size_t)bb * CM * Pn;
  const int m = mt * 16 + l16;
  const int n0 = p0 + l16;
  for (int k0 = 0; k0 < CM; k0 += 4) {
    const int k = k0 + half * 2;
    // contiguous A pair: one 8-byte load
    const v2f a = *(const v2f*)(WTi + ((size_t)(k0 >> 2) * OP + m) * 4 + half * 2);
    const float* xr0 = xb + (size_t)k * Pn;
    const float* xr1 = xb + (size_t)(k + 1) * Pn;
#pragma unroll
    for (int nt = 0; nt < 4; ++nt) {
      v2f b;
      b.x = xr0[n0 + nt * 16];
      b.y = xr1[n0 + nt * 16];
      acc[nt] = __builtin_amdgcn_wmma_f32_16x16x4_f32(false, a, false, b,
                                                      (short)0, acc[nt],
                                                      false, false);
    }
  }
}

// ---------------------------------------------------------------------------
// W (536x256) -> WTi packed [k/4][m][4], zero-padded rows 536..543
// ---------------------------------------------------------------------------
__global__ void build_wt_kernel(const float* __restrict__ W,
                                float* __restrict__ WTi) {
  int i = blockIdx.x * blockDim.x + threadIdx.x;   // over OP*CM
  if (i >= OP * CM) return;
  // i = ((k>>2)*OP + m)*4 + (k&3)
  int jj = i & 3;
  int t  = i >> 2;
  int m  = t % OP;
  int kg = t / OP;
  int k  = kg * 4 + jj;
  WTi[i] = (m < Oc) ? W[(size_t)m * CM + k] : 0.f;
}

// ---------------------------------------------------------------------------
// Pass 1: WMMA GEMM accumulating per-channel sum / sum-of-squares of kv.
// A wave owns a full M-row (fixed channels) across all 4 N-tiles, so
// partials accumulate in registers and reduce via a single ds_add_f32.
// ---------------------------------------------------------------------------
__global__ void __launch_bounds__(256)
gemm_stats_kernel(const float* __restrict__ X, const float* __restrict__ WTi,
                  float* __restrict__ chSum, float* __restrict__ chSq) {
  __shared__ float sSum[OP];
  __shared__ float sSq[OP];
  const int wg = blockIdx.x;
  const int bb = wg >> 8;            // /PT
  const int p0 = (wg & (PT - 1)) * NT;
  for (int i = threadIdx.x; i < OP; i += blockDim.x) { sSum[i] = 0.f; sSq[i] = 0.f; }
  __syncthreads();

  const int wv = threadIdx.x >> 5;
  const int lane = threadIdx.x & 31;
  const int half = lane >> 4;

  for (int mt = wv; mt < MT; mt += 8) {
    v8f acc[4] = {{}, {}, {}, {}};
    wmma_row(X, WTi, bb, mt, p0, acc);
#pragma unroll
    for (int r = 0; r < 8; ++r) {
      float s = 0.f, q = 0.f;
#pragma unroll
      for (int nt = 0; nt < 4; ++nt) {
        const float v = acc[nt][r];
        s += v;
        q += v * v;
      }
      const int row = mt * 16 + r + half * 8;
      atomicAdd(&sSum[row], s);   // ds_add_f32 (16-way, hidden by 8 waves)
      atomicAdd(&sSq[row], q);
    }
  }
  __syncthreads();
  for (int i = threadIdx.x; i < Oc; i += blockDim.x) {
    atomicAdd(&chSum[i], sSum[i]);
    atomicAdd(&chSq[i], sSq[i]);
  }
}

// ---------------------------------------------------------------------------
// Fold BN stats + gamma/beta into per-channel scale (A) and shift (C).
// ---------------------------------------------------------------------------
__global__ void finalize_kernel(const float* __restrict__ chSum,
                                const float* __restrict__ chSq,
                                const float* __restrict__ kg,
                                const float* __restrict__ kb,
                                const float* __restrict__ vg,
                                const float* __restrict__ vb,
                                float* __restrict__ nrmA,
                                float* __restrict__ nrmC) {
  int i = blockIdx.x * blockDim.x + threadIdx.x;
  if (i >= OP) return;
  if (i >= Oc) { nrmA[i] = 0.f; nrmC[i] = 0.f; return; }
  const float invN = 1.0f / (float)(Bc * Pn);
  float mean = chSum[i] * invN;
  float var  = chSq[i] * invN - mean * mean;
  float rstd = rsqrtf(var + 1e-5f);
  float g = (i < Hh * 3) ? kg[i] : vg[i - Hh * 3];
  float b = (i < Hh * 3) ? kb[i] : vb[i - Hh * 3];
  float a = rstd * g;
  nrmA[i] = a;
  nrmC[i] = b - mean * a;
}

// ---------------------------------------------------------------------------
// Pass 2: WMMA GEMM -> normalize -> LDS column store -> keys/tanh/splat.
// grid layout (b, h, cell, f): f contiguous -> coalesced wave atomics.
// ---------------------------------------------------------------------------
__global__ void __launch_bounds__(256)
gemm_splat_kernel(const float* __restrict__ X, const float* __restrict__ WTi,
                  const float* __restrict__ nrmA, const float* __restrict__ nrmC,
                  const float* __restrict__ orig, const float* __restrict__ T,
                  float* __restrict__ grid, float* __restrict__ keyStats) {
  extern __shared__ float smem[];
  float* kvn   = smem;                 // OP * LDSW
  float* kstat = smem + OP * LDSW;     // [2]

  const int wg = blockIdx.x;
  const int bb = wg >> 8;
  const int p0 = (wg & (PT - 1)) * NT;
  const int tid = threadIdx.x;
  const int wv = tid >> 5, lane = tid & 31;
  const int half = lane >> 4, l16 = lane & 15;

  if (tid < 2) kstat[tid] = 0.f;

  // ---- phase 1: GEMM + normalize into LDS (nrm coeffs reused across N) ----
  for (int mt = wv; mt < MT; mt += 8) {
    v8f acc[4] = {{}, {}, {}, {}};
    wmma_row(X, WTi, bb, mt, p0, acc);
#pragma unroll
    for (int r = 0; r < 8; ++r) {
      const int row = mt * 16 + r + half * 8;
      const float A = nrmA[row];
      const float C = nrmC[row];
      float* dst = &kvn[row * LDSW + l16];
#pragma unroll
      for (int nt = 0; nt < 4; ++nt)
        dst[nt * 16] = acc[nt][r] * A + C;
    }
  }
  __syncthreads();

  // ---- phase 2: each wave consumes 8 full columns (points) ----
  float ksum = 0.f, ksq = 0.f;
  for (int i = 0; i < 8; ++i) {
    const int ploc = wv * 8 + i;
    const int pg = p0 + ploc;
    const float o0 = orig[((size_t)bb * 3 + 0) * Pn + pg];
    const float o1 = orig[((size_t)bb * 3 + 1) * Pn + pg];
    const float o2 = orig[((size_t)bb * 3 + 2) * Pn + pg];
#pragma unroll
    for (int h = 0; h < Hh; ++h) {
      const float p0c = o0 + kvn[(h * 3 + 0) * LDSW + ploc];
      const float p1c = o1 + kvn[(h * 3 + 1) * LDSW + ploc];
      const float p2c = o2 + kvn[(h * 3 + 2) * LDSW + ploc];
      const float* Th = T + h * 6;
      const float tk0 = Th[0] * p0c + Th[1] * p1c + Th[2] * p2c;
      const float tk1 = Th[3] * p0c + Th[4] * p1c + Th[5] * p2c;
      ksum += tk0 + tk1;
      ksq  += tk0 * tk0 + tk1 * tk1;
      const float pos0 = (tanhf(tk0) + 1.f) * 0.5f * (float)(Gg - 1);
      const float pos1 = (tanhf(tk1) + 1.f) * 0.5f * (float)(Gg - 1);
      const float b0f = fminf(fmaxf(floorf(pos0), 0.f), (float)(Gg - 2));
      const float b1f = fminf(fmaxf(floorf(pos1), 0.f), (float)(Gg - 2));
      const float fx = pos0 - b0f, fy = pos1 - b1f;
      const int bx = (int)b0f, by = (int)b1f;

      // feature values for this (b,h,point): conflict-free stride-65 LDS reads
      const float v0 = kvn[(Hh * 3 + h * Ff + lane) * LDSW + ploc];
      const float v1 = kvn[(Hh * 3 + h * Ff + 32 + lane) * LDSW + ploc];
      float* gbase = grid + ((size_t)(bb * Hh + h) * (Gg * Gg)) * Ff;
#pragma unroll
      for (int cx = 0; cx < 2; ++cx) {
        const float wx = cx ? fx : 1.f - fx;
#pragma unroll
        for (int cy = 0; cy < 2; ++cy) {
          const float w = wx * (cy ? fy : 1.f - fy);
          const int cell = (bx + cx) * Gg + (by + cy);
          float* gc = gbase + (size_t)cell * Ff;
          atomicAdd(gc + lane,      w * v0);   // coalesced 128B wave atomic
          atomicAdd(gc + lane + 32, w * v1);
        }
      }
    }
  }
  if (lane == 0) { atomicAdd(&kstat[0], ksum); atomicAdd(&kstat[1], ksq); }
  __syncthreads();
  if (tid == 0) { atomicAdd(&keyStats[0], kstat[0]); atomicAdd(&keyStats[1], kstat[1]); }
}

// ---------------------------------------------------------------------------
// grid (b,h,cell,f) -> out (b, h*F+f, cell), fused occupancy count.
// ---------------------------------------------------------------------------
__global__ void transpose_occ_kernel(const float* __restrict__ grid,
                                     float* __restrict__ out,
                                     unsigned* __restrict__ occ) {
  __shared__ unsigned cnt;
  if (threadIdx.x == 0) cnt = 0u;
  __syncthreads();
  const size_t o = (size_t)blockIdx.x * blockDim.x + threadIdx.x;
  const unsigned bh = (unsigned)(o >> 18);
  const unsigned f  = (unsigned)(o >> 12) & 63u;
  const unsigned cell = (unsigned)o & 4095u;
  const float v = grid[((size_t)bh * 4096 + cell) * Ff + f];
  out[o] = v;
  const unsigned nz = (fabsf(v) > 1e-9f) ? 1u : 0u;
  unsigned long long m = __ballot(nz);
  if ((threadIdx.x & 31) == 0) atomicAdd(&cnt, (unsigned)__popcll(m));
  __syncthreads();
  if (threadIdx.x == 0) atomicAdd(occ, cnt);
}

__global__ void scalars_kernel(const unsigned* __restrict__ occ,
                               const float* __restrict__ keyStats,
                               float* __restrict__ out) {
  if (threadIdx.x == 0 && blockIdx.x == 0) {
    out[16777216] = (float)(*occ) / (float)(Bc * Ff * Hh);
    const float N = (float)(Bc * Hh * 2 * Pn);      // 2,097,152 keys
    const float S1 = keyStats[0], S2 = keyStats[1];
    const float mean = S1 / N;
    out[16777217] = mean;
    out[16777218] = (S2 - S1 * mean) / (N - 1.0f);  // ddof = 1
  }
}

// ---------------------------------------------------------------------------
// Workspace layout (floats):
//   [0, 16777216)        grid (B,H,4096,F)
//   16777216 .. +536     chSum
//   16777752 .. +536     chSq
//   16778288 .. +544     nrmA
//   16778832 .. +544     nrmC
//   16779376 .. +2       keyStats
//   16779378             occ (unsigned)
//   16779392 .. +139264  WTi (packed 256 x 544)
// ---------------------------------------------------------------------------
extern "C" void kernel_launch(void* const* d_in, const int* in_sizes, int n_in,
                              void* d_out, int out_size, void* d_ws, size_t ws_size,
                              hipStream_t stream) {
  const float* X    = (const float*)d_in[0];
  const float* orig = (const float*)d_in[1];
  const float* W    = (const float*)d_in[2];
  const float* kg   = (const float*)d_in[3];
  const float* kb   = (const float*)d_in[4];
  const float* vg   = (const float*)d_in[5];
  const float* vb   = (const float*)d_in[6];
  const float* T    = (const float*)d_in[7];
  float* out = (float*)d_out;

  float* ws      = (float*)d_ws;
  float* grid    = ws;
  float* chSum   = ws + 16777216;
  float* chSq    = ws + 16777752;
  float* nrmA    = ws + 16778288;
  float* nrmC    = ws + 16778832;
  float* keySt   = ws + 16779376;
  unsigned* occ  = (unsigned*)(ws + 16779378);
  float* WTi     = ws + 16779392;

  // zero grid + stats (WTi fully overwritten by build kernel)
  hipMemsetAsync(d_ws, 0, (size_t)16779392 * sizeof(float), stream);

  build_wt_kernel<<<(OP * CM + 255) / 256, 256, 0, stream>>>(W, WTi);

  gemm_stats_kernel<<<Bc * PT, 256, 0, stream>>>(X, WTi, chSum, chSq);

  finalize_kernel<<<(OP + 255) / 256, 256, 0, stream>>>(chSum, chSq, kg, kb, vg, vb,
                                                        nrmA, nrmC);

  const size_t smem = (size_t)(OP * LDSW) * sizeof(float) + 16;
  hipFuncSetAttribute((const void*)gemm_splat_kernel,
                      hipFuncAttributeMaxDynamicSharedMemorySize, (int)smem);
  gemm_splat_kernel<<<Bc * PT, 256, smem, stream>>>(X, WTi, nrmA, nrmC, orig, T,
                                                    grid, keySt);

  transpose_occ_kernel<<<(Bc * Hh * Ff * Gg * Gg) / 256, 256, 0, stream>>>(grid, out, occ);

  scalars_kernel<<<1, 32, 0, stream>>>(occ, keySt, out);
}